// ChamferDistanceLoss_5025111736600
// MI455X (gfx1250) — compile-verified
//
#include <hip/hip_runtime.h>

typedef __attribute__((ext_vector_type(2))) float v2f;
typedef __attribute__((ext_vector_type(8))) float v8f;

#define NPTS      10000
#define NPTS_PAD  10240        // 640 tiles of 16  (divisible by 8 waves & 16-tile chunks)
#define NTILES_PAD 640
#define CHUNK_T   16           // column tiles staged per LDS chunk (256 points, 4KB)
#define NCHUNKS   (NTILES_PAD / CHUNK_T)   // 40
#define VOX_PER   (64*64*64)   // 262144
#define NCHK      256          // voxel chunks per (batch,set), 1024 voxels each
#define PADC      1.0e15f
#define PADN      3.0e30f

// ---------------------------------------------------------------------------
// Phase 1a: per-chunk active counts.  grid (256, 8) x 1024 threads
// ---------------------------------------------------------------------------
__global__ __launch_bounds__(1024)
void count_active(const float* __restrict__ pred,
                  const float* __restrict__ target,
                  int* __restrict__ chunk_counts) {
  const int chunk = blockIdx.x;            // 0..255
  const int bs    = blockIdx.y;            // 0..7 : batch*2 + set
  const float* vox = ((bs & 1) == 0 ? pred : target) + (bs >> 1) * VOX_PER;

  const int tid  = threadIdx.x;
  const int lane = tid & 31;
  const int wid  = tid >> 5;
  __shared__ int wsum[32];

  const bool act = vox[chunk * 1024 + tid] > 0.5f;
  const unsigned m = (unsigned)__ballot(act);
  if (lane == 0) wsum[wid] = __popc(m);
  __syncthreads();
  if (tid < 32) {
    int v = wsum[tid];
    v += __shfl_xor(v, 16, 32);
    v += __shfl_xor(v, 8, 32);
    v += __shfl_xor(v, 4, 32);
    v += __shfl_xor(v, 2, 32);
    v += __shfl_xor(v, 1, 32);
    if (tid == 0) chunk_counts[bs * NCHK + chunk] = v;
  }
}

// ---------------------------------------------------------------------------
// Phase 1b: exclusive scan of 256 chunk counts per (batch,set). 1 block x 256
// ---------------------------------------------------------------------------
__global__ __launch_bounds__(256)
void scan_counts(const int* __restrict__ chunk_counts,
                 int* __restrict__ chunk_bases,
                 int* __restrict__ counts_nv) {
  __shared__ int buf[256];
  const int tid = threadIdx.x;
  for (int s = 0; s < 8; ++s) {
    const int v = chunk_counts[s * NCHK + tid];
    buf[tid] = v;
    __syncthreads();
    for (int off = 1; off < 256; off <<= 1) {          // Hillis-Steele inclusive
      const int t = (tid >= off) ? buf[tid - off] : 0;
      __syncthreads();
      buf[tid] += t;
      __syncthreads();
    }
    const int incl = buf[tid];
    chunk_bases[s * NCHK + tid] = incl - v;            // exclusive prefix
    if (tid == 255) counts_nv[s] = incl > NPTS ? NPTS : incl;
    __syncthreads();
  }
}

// ---------------------------------------------------------------------------
// Phase 1c: parallel ordered scatter -> B-form points. grid (256, 8) x 1024
// bform[bs*NPTS_PAD + i] = (x, y, z, x^2+y^2+z^2)
// ---------------------------------------------------------------------------
__global__ __launch_bounds__(1024)
void scatter_points(const float* __restrict__ pred,
                    const float* __restrict__ target,
                    const int* __restrict__ chunk_bases,
                    float4* __restrict__ bform) {
  const int chunk = blockIdx.x;
  const int bs    = blockIdx.y;
  const float* vox = ((bs & 1) == 0 ? pred : target) + (bs >> 1) * VOX_PER;

  const int tid  = threadIdx.x;
  const int lane = tid & 31;
  const int wid  = tid >> 5;
  __shared__ int wsum[32];
  __shared__ int wpre[32];

  const int idx  = chunk * 1024 + tid;
  const bool act = vox[idx] > 0.5f;
  const unsigned m = (unsigned)__ballot(act);
  const int rank = __popc(m & ((1u << lane) - 1u));
  if (lane == 0) wsum[wid] = __popc(m);
  __syncthreads();
  if (tid < 32) {                                      // wave 0: scan 32 wave sums
    int v = wsum[tid];
    #pragma unroll
    for (int off = 1; off < 32; off <<= 1) {
      int u = __shfl_up(v, off, 32);
      if (tid >= off) v += u;
    }
    wpre[tid] = v - wsum[tid];
  }
  __syncthreads();

  const int pos = chunk_bases[bs * NCHK + chunk] + wpre[wid] + rank;
  if (act && pos < NPTS) {
    const int d = idx >> 12;
    const int h = (idx >> 6) & 63;
    const int w = idx & 63;
    const float x = (float)d / 63.0f * 2.0f - 1.0f;
    const float y = (float)h / 63.0f * 2.0f - 1.0f;
    const float z = (float)w / 63.0f * 2.0f - 1.0f;
    bform[bs * NPTS_PAD + pos] = make_float4(x, y, z, x*x + y*y + z*z);
  }
}

// ---------------------------------------------------------------------------
// Phase 1d: fill pads [nv, NPTS_PAD).  grid (10, 8) x 1024
// ---------------------------------------------------------------------------
__global__ __launch_bounds__(1024)
void pad_points(const int* __restrict__ counts_nv, float4* __restrict__ bform) {
  const int bs = blockIdx.y;
  const int i  = blockIdx.x * 1024 + threadIdx.x;      // 0..10239
  if (i >= counts_nv[bs] && i < NPTS_PAD)
    bform[bs * NPTS_PAD + i] = make_float4(PADC, PADC, PADC, PADN);
}

// ---------------------------------------------------------------------------
// Phase 2: chamfer tiles.  grid (80, 4, 2) x 256 (8 waves).
// Each wave owns one 16-row tile; B tiles staged chunk-wise in LDS via
// async global->LDS copies, consumed by all 8 waves (8x L2-traffic cut).
// q = n2 - 2*dot via WMMA f32 16x16x4 with A=[-2x,-2y,-2z,1], B=[x,y,z,n2].
// ---------------------------------------------------------------------------
__global__ __launch_bounds__(256)
void chamfer_tiles(const float4* __restrict__ bform,
                   const int* __restrict__ counts_nv,
                   float* __restrict__ partials) {
  __shared__ float4 sB[CHUNK_T * 16];                  // 256 points = 4KB

  const int batch = blockIdx.y;
  const int dir   = blockIdx.z;                        // 0: pred->target, 1: reverse
  const int asel  = dir;
  const int bsel  = 1 - dir;
  const float4* Af = bform + (batch * 2 + asel) * NPTS_PAD;
  const float4* Bf = bform + (batch * 2 + bsel) * NPTS_PAD;
  const int nvA = counts_nv[batch * 2 + asel];

  const int tid  = threadIdx.x;
  const int lane = tid & 31;
  const int wid  = tid >> 5;
  const int rt   = blockIdx.x * 8 + wid;               // 0..639 (pads give 0)
  const bool hi  = lane >= 16;
  const int l15  = lane & 15;

  // A tile (ISA 16x4 f32 layout: lane=M, VGPR0/1 = K{0,1} lo-half / K{2,3} hi-half)
  const float4 pa = Af[rt * 16 + l15];
  v2f a;
  a.x = hi ? (-2.0f * pa.z) : (-2.0f * pa.x);
  a.y = hi ? 1.0f           : (-2.0f * pa.y);

  v8f rmin;
  #pragma unroll
  for (int r = 0; r < 8; ++r) rmin[r] = 3.0e38f;

  const unsigned ldsaddr = (unsigned)(size_t)&sB[tid]; // low 32 bits = LDS offset
  for (int ch = 0; ch < NCHUNKS; ++ch) {
    // async copy: 256 threads each move one float4 of the B chunk into LDS
    const unsigned long long gaddr =
        (unsigned long long)(size_t)&Bf[ch * (CHUNK_T * 16) + tid];
    asm volatile("global_load_async_to_lds_b128 %0, %1, off"
                 :: "v"(ldsaddr), "v"(gaddr) : "memory");
    asm volatile("s_wait_asynccnt 0" ::: "memory");
    __syncthreads();                                   // LDS chunk visible to all waves

    #pragma unroll 4
    for (int t = 0; t < CHUNK_T; ++t) {
      const float4 pb = sB[t * 16 + l15];              // ds_load_b128
      v2f b;
      b.x = hi ? pb.z : pb.x;                          // K0 / K2
      b.y = hi ? pb.w : pb.y;                          // K1 / K3 (carries n2)
      v8f c = {};
      v8f dacc = __builtin_amdgcn_wmma_f32_16x16x4_f32(
          false, a, false, b, (short)0, c, false, false);
      #pragma unroll
      for (int r = 0; r < 8; ++r) rmin[r] = fminf(rmin[r], dacc[r]);
    }
    __syncthreads();                                   // before chunk overwrite
  }

  // min across the 16 lanes of each half (rows live per-VGPR per-half)
  #pragma unroll
  for (int r = 0; r < 8; ++r) {
    float v = rmin[r];
    v = fminf(v, __shfl_xor(v, 1, 32));
    v = fminf(v, __shfl_xor(v, 2, 32));
    v = fminf(v, __shfl_xor(v, 4, 32));
    v = fminf(v, __shfl_xor(v, 8, 32));
    rmin[r] = v;
  }

  const int add8 = hi ? 8 : 0;
  float part = 0.0f;
  #pragma unroll
  for (int r = 0; r < 8; ++r) {
    const int lr = r + add8;                           // local row 0..15
    const float n1 = __shfl(pa.w, lr, 32);             // ||p1||^2 of that row
    const float dmin = fmaxf(n1 + rmin[r], 0.0f);      // clamp commutes with min
    if (rt * 16 + lr < nvA) part += dmin;
  }
  const float tot = part + __shfl_xor(part, 16, 32);
  if (lane == 0) partials[(dir * 4 + batch) * NTILES_PAD + rt] = tot;
}

// ---------------------------------------------------------------------------
// Phase 3: deterministic final reduction + batch averaging
// ---------------------------------------------------------------------------
__global__ __launch_bounds__(256)
void finalize(const float* __restrict__ partials,
              const int* __restrict__ counts_nv,
              float* __restrict__ out) {
  __shared__ float red[256];
  __shared__ float gsum[8];
  const int tid = threadIdx.x;

  for (int g = 0; g < 8; ++g) {
    float s = 0.0f;
    for (int i = tid; i < NTILES_PAD; i += 256) s += partials[g * NTILES_PAD + i];
    red[tid] = s;
    __syncthreads();
    for (int st = 128; st > 0; st >>= 1) {
      if (tid < st) red[tid] += red[tid + st];
      __syncthreads();
    }
    if (tid == 0) gsum[g] = red[0];
    __syncthreads();
  }

  if (tid == 0) {
    float total = 0.0f, nvalid = 0.0f;
    for (int b = 0; b < 4; ++b) {
      const int n1 = counts_nv[b * 2 + 0];
      const int n2 = counts_nv[b * 2 + 1];
      const float c1 = fmaxf((float)n1, 1.0f);
      const float c2 = fmaxf((float)n2, 1.0f);
      const float cd = gsum[0 * 4 + b] / c1 + gsum[1 * 4 + b] / c2;
      if (n1 > 0 && n2 > 0) { total += cd; nvalid += 1.0f; }
    }
    out[0] = (nvalid > 0.0f) ? (total / fmaxf(nvalid, 1.0f)) : 0.0f;
  }
}

// ---------------------------------------------------------------------------
extern "C" void kernel_launch(void* const* d_in, const int* in_sizes, int n_in,
                              void* d_out, int out_size, void* d_ws, size_t ws_size,
                              hipStream_t stream) {
  const float* pred   = (const float*)d_in[0];   // (4,1,64,64,64) f32
  const float* target = (const float*)d_in[1];   // (4,1,64,64,64) f32

  // workspace layout (~1.35 MB)
  int*    counts_nv    = (int*)d_ws;                          // 8 ints
  int*    chunk_counts = (int*)((char*)d_ws + 64);            // 8*256 ints
  int*    chunk_bases  = (int*)((char*)d_ws + 8320);          // 8*256 ints
  float*  partials     = (float*)((char*)d_ws + 16640);       // 8*640 floats
  float4* bform        = (float4*)((char*)d_ws + 40960);      // 8*10240 float4

  dim3 gc(NCHK, 8);
  count_active<<<gc, 1024, 0, stream>>>(pred, target, chunk_counts);
  scan_counts<<<1, 256, 0, stream>>>(chunk_counts, chunk_bases, counts_nv);
  scatter_points<<<gc, 1024, 0, stream>>>(pred, target, chunk_bases, bform);
  dim3 gp(NPTS_PAD / 1024, 8);
  pad_points<<<gp, 1024, 0, stream>>>(counts_nv, bform);

  dim3 g2(NTILES_PAD / 8, 4, 2);
  chamfer_tiles<<<g2, 256, 0, stream>>>(bform, counts_nv, partials);

  finalize<<<1, 256, 0, stream>>>(partials, counts_nv, (float*)d_out);
}